// GAT_43980465111678
// MI455X (gfx1250) — compile-verified
//
#include <hip/hip_runtime.h>
#include <hip/hip_bf16.h>

typedef __attribute__((ext_vector_type(2))) float v2f;
typedef __attribute__((ext_vector_type(8))) float v8f;

#define HID 64
#define HEADS 4
#define CCH 16
#define NEG_SLOPE 0.2f
#define LN_EPS 1e-5f

// ---------------------------------------------------------------------------
// WMMA GEMM: Z[N,64] = A[N,K] @ W[64,K]^T   (f32, V_WMMA_F32_16X16X4_F32)
// One block = 16 rows of Z. 4 waves; wave w computes the 16x16 tile for
// columns [16w, 16w+16) == head w. A tile staged into LDS (coalesced, padded).
//
// f32 WMMA operand layout (ISA 7.12.2):
//   A 16x4:  lanes 0-15 hold row M=lane, {v0,v1} = K,K+1 ; lanes 16-31 = K+2,K+3
//   B 4x16:  mirrored (lanes 0-15 rows K,K+1 ; lanes 16-31 rows K+2,K+3), N=lane&15
//   C/D 16x16: VGPR r -> (M=r, N=lane) lanes 0-15 ; (M=r+8, N=lane-16) lanes 16-31
// ---------------------------------------------------------------------------
template <int K>
__global__ void gat_gemm_wmma(const float* __restrict__ A,
                              const float* __restrict__ W,
                              float* __restrict__ Z, int N) {
  __shared__ float As[16 * (K + 4)];
  const int tid  = threadIdx.x;
  const int wave = tid >> 5;
  const int lane = tid & 31;
  const int half = lane >> 4;
  const int l16  = lane & 15;
  const int row0 = blockIdx.x * 16;
  const int col0 = wave * 16;

  // Stage A tile (16 rows x K) into LDS, fully coalesced, padded stride.
  for (int i = tid; i < 16 * K; i += 128) {
    int r = i / K, c = i % K;
    int row = row0 + r; if (row >= N) row = N - 1;
    As[r * (K + 4) + c] = A[(size_t)row * K + c];
  }
  __syncthreads();

  const float* arow = As + l16 * (K + 4);
  const float* wrow = W + (size_t)(col0 + l16) * K;

  v8f acc = {};
#pragma unroll
  for (int k = 0; k < K; k += 4) {
    const int kk = k + 2 * half;
    v2f a; a[0] = arow[kk]; a[1] = arow[kk + 1];
    v2f b; b[0] = wrow[kk]; b[1] = wrow[kk + 1];
    acc = __builtin_amdgcn_wmma_f32_16x16x4_f32(
        /*neg_a=*/false, a, /*neg_b=*/false, b,
        /*c_mod=*/(short)0, acc, /*reuse_a=*/false, /*reuse_b=*/false);
  }

#pragma unroll
  for (int r = 0; r < 8; ++r) {
    int rr = row0 + r + 8 * half;
    if (rr < N) Z[(size_t)rr * HID + col0 + l16] = acc[r];
  }
}

// ---------------------------------------------------------------------------
// Per-(node,head) attention logits: e_src/e_dst = dot(z[n, h*16:+16], a[h])
// ---------------------------------------------------------------------------
__global__ void node_scores(const float* __restrict__ z,
                            const float* __restrict__ asrc,
                            const float* __restrict__ adst,
                            float* __restrict__ esrc,
                            float* __restrict__ edst, int N) {
  int t = blockIdx.x * blockDim.x + threadIdx.x;
  if (t >= N * HEADS) return;
  int n = t >> 2, hd = t & 3;
  const float* zp = z + (size_t)n * HID + hd * CCH;
  const float* as = asrc + hd * CCH;
  const float* ad = adst + hd * CCH;
  float s = 0.f, d = 0.f;
#pragma unroll
  for (int c = 0; c < CCH; ++c) { float zv = zp[c]; s += zv * as[c]; d += zv * ad[c]; }
  esrc[t] = s; edst[t] = d;
}

__global__ void init_buffers(float* __restrict__ agg, float* __restrict__ m,
                             float* __restrict__ den, int N) {
  size_t t = (size_t)blockIdx.x * blockDim.x + threadIdx.x;
  if (t < (size_t)N * HID) agg[t] = 0.f;
  if (t < (size_t)N * HEADS) { m[t] = -INFINITY; den[t] = 0.f; }
}

// Sign-aware float atomic max via int/uint punning (valid with -inf init).
__device__ __forceinline__ void atomicMaxF(float* addr, float v) {
  if (v >= 0.f) atomicMax((int*)addr, __float_as_int(v));
  else          atomicMin((unsigned int*)addr, __float_as_uint(v));
}

__device__ __forceinline__ void edge_pair(const int* __restrict__ ei, int e,
                                          int E, int& s, int& d) {
  if (e < E) { s = ei[e]; d = ei[E + e]; }
  else       { s = d = e - E; }  // self loop
}

__device__ __forceinline__ float edge_logit(const float* __restrict__ esrc,
                                            const float* __restrict__ edst,
                                            int s, int d, int hd) {
  float v = esrc[s * HEADS + hd] + edst[d * HEADS + hd];
  return (v > 0.f) ? v : NEG_SLOPE * v;  // leaky relu
}

__global__ void edge_max(const int* __restrict__ ei,
                         const float* __restrict__ esrc,
                         const float* __restrict__ edst,
                         float* __restrict__ m, int E, int Etot) {
  int t = blockIdx.x * blockDim.x + threadIdx.x;
  if (t >= Etot * HEADS) return;
  int e = t >> 2, hd = t & 3, s, d;
  edge_pair(ei, e, E, s, d);
  atomicMaxF(&m[d * HEADS + hd], edge_logit(esrc, edst, s, d, hd));
}

__global__ void edge_expsum(const int* __restrict__ ei,
                            const float* __restrict__ esrc,
                            const float* __restrict__ edst,
                            const float* __restrict__ m,
                            float* __restrict__ den, int E, int Etot) {
  int t = blockIdx.x * blockDim.x + threadIdx.x;
  if (t >= Etot * HEADS) return;
  int e = t >> 2, hd = t & 3, s, d;
  edge_pair(ei, e, E, s, d);
  float v = edge_logit(esrc, edst, s, d, hd);
  atomicAdd(&den[d * HEADS + hd], __expf(v - m[d * HEADS + hd]));
}

__global__ void edge_aggregate(const int* __restrict__ ei,
                               const float* __restrict__ esrc,
                               const float* __restrict__ edst,
                               const float* __restrict__ m,
                               const float* __restrict__ den,
                               const float* __restrict__ z,
                               float* __restrict__ agg, int E, int Etot) {
  int t = blockIdx.x * blockDim.x + threadIdx.x;
  if (t >= Etot * HEADS) return;
  int e = t >> 2, hd = t & 3, s, d;
  edge_pair(ei, e, E, s, d);
  float v = edge_logit(esrc, edst, s, d, hd);
  float alpha = __expf(v - m[d * HEADS + hd]) / den[d * HEADS + hd];
  const float4* zp = (const float4*)(z + (size_t)s * HID + hd * CCH);
  float* ap = agg + (size_t)d * HID + hd * CCH;
#pragma unroll
  for (int i = 0; i < 4; ++i) {
    float4 zv = zp[i];
    atomicAdd(ap + 4 * i + 0, zv.x * alpha);
    atomicAdd(ap + 4 * i + 1, zv.y * alpha);
    atomicAdd(ap + 4 * i + 2, zv.z * alpha);
    atomicAdd(ap + 4 * i + 3, zv.w * alpha);
  }
}

// ---------------------------------------------------------------------------
// Per-node epilogue: bias + LayerNorm + ReLU (+ optional residual).
// One wave (32 lanes) per node, 2 channels per lane, shfl_xor reductions.
// ---------------------------------------------------------------------------
__global__ void finish_node(const float* __restrict__ agg,
                            const float* __restrict__ bias,
                            const float* __restrict__ g,
                            const float* __restrict__ be,
                            const float* __restrict__ resid,
                            float* __restrict__ out, int N) {
  int wave = threadIdx.x >> 5, lane = threadIdx.x & 31;
  int n = blockIdx.x * (blockDim.x >> 5) + wave;
  if (n >= N) return;
  size_t base = (size_t)n * HID + lane * 2;
  float2 v  = *(const float2*)(agg + base);
  float2 bb = *(const float2*)(bias + lane * 2);
  float x0 = v.x + bb.x, x1 = v.y + bb.y;
  float s = x0 + x1;
#pragma unroll
  for (int o = 16; o > 0; o >>= 1) s += __shfl_xor(s, o);
  float mu = s * (1.0f / HID);
  float d0 = x0 - mu, d1 = x1 - mu;
  float q = d0 * d0 + d1 * d1;
#pragma unroll
  for (int o = 16; o > 0; o >>= 1) q += __shfl_xor(q, o);
  float rstd = rsqrtf(q * (1.0f / HID) + LN_EPS);
  float2 gg  = *(const float2*)(g + lane * 2);
  float2 bbe = *(const float2*)(be + lane * 2);
  float y0 = fmaxf(d0 * rstd * gg.x + bbe.x, 0.f);
  float y1 = fmaxf(d1 * rstd * gg.y + bbe.y, 0.f);
  if (resid) { y0 += resid[base]; y1 += resid[base + 1]; }
  float2 r; r.x = y0; r.y = y1;
  *(float2*)(out + base) = r;
}

// coords[n,0:2] = h[n,:] @ Wout^T + bout  (wave per node)
__global__ void out_head(const float* __restrict__ h,
                         const float* __restrict__ Wout,
                         const float* __restrict__ bout,
                         float* __restrict__ out, int N) {
  int wave = threadIdx.x >> 5, lane = threadIdx.x & 31;
  int n = blockIdx.x * (blockDim.x >> 5) + wave;
  if (n >= N) return;
  float2 hv = *(const float2*)(h + (size_t)n * HID + lane * 2);
  float2 w0 = *(const float2*)(Wout + lane * 2);
  float2 w1 = *(const float2*)(Wout + HID + lane * 2);
  float p0 = hv.x * w0.x + hv.y * w0.y;
  float p1 = hv.x * w1.x + hv.y * w1.y;
#pragma unroll
  for (int o = 16; o > 0; o >>= 1) { p0 += __shfl_xor(p0, o); p1 += __shfl_xor(p1, o); }
  if (lane == 0) { out[(size_t)n * 2] = p0 + bout[0]; out[(size_t)n * 2 + 1] = p1 + bout[1]; }
}

extern "C" void kernel_launch(void* const* d_in, const int* in_sizes, int n_in,
                              void* d_out, int out_size, void* d_ws, size_t ws_size,
                              hipStream_t stream) {
  const float* x     = (const float*)d_in[0];
  const int*   ei    = (const int*)d_in[1];
  const float* W0    = (const float*)d_in[2];
  const float* asrc0 = (const float*)d_in[3];
  const float* adst0 = (const float*)d_in[4];
  const float* b0    = (const float*)d_in[5];
  const float* g0    = (const float*)d_in[6];
  const float* be0   = (const float*)d_in[7];
  const float* W1    = (const float*)d_in[8];
  const float* asrc1 = (const float*)d_in[9];
  const float* adst1 = (const float*)d_in[10];
  const float* b1    = (const float*)d_in[11];
  const float* g1    = (const float*)d_in[12];
  const float* be1   = (const float*)d_in[13];
  const float* Wout  = (const float*)d_in[14];
  const float* bout  = (const float*)d_in[15];

  const int N = in_sizes[0] / 128;
  const int E = in_sizes[1] / 2;
  const int Etot = E + N;  // with self loops

  // Workspace carve-up (floats): z | agg | h | esrc | edst | m | den
  float* ws   = (float*)d_ws;
  float* z    = ws;
  float* agg  = z   + (size_t)N * HID;
  float* h    = agg + (size_t)N * HID;
  float* esrc = h   + (size_t)N * HID;
  float* edst = esrc + (size_t)N * HEADS;
  float* m    = edst + (size_t)N * HEADS;
  float* den  = m    + (size_t)N * HEADS;

  const int gemmGrid = (N + 15) / 16;
  const int nsGrid   = (N * HEADS + 255) / 256;
  const int initGrid = (int)(((size_t)N * HID + 255) / 256);
  const int edgeGrid = (Etot * HEADS + 255) / 256;
  const int nodeGrid = (N + 7) / 8;  // 8 waves/block, wave per node

  // ---- Layer 0 ----
  gat_gemm_wmma<128><<<gemmGrid, 128, 0, stream>>>(x, W0, z, N);
  node_scores<<<nsGrid, 256, 0, stream>>>(z, asrc0, adst0, esrc, edst, N);
  init_buffers<<<initGrid, 256, 0, stream>>>(agg, m, den, N);
  edge_max<<<edgeGrid, 256, 0, stream>>>(ei, esrc, edst, m, E, Etot);
  edge_expsum<<<edgeGrid, 256, 0, stream>>>(ei, esrc, edst, m, den, E, Etot);
  edge_aggregate<<<edgeGrid, 256, 0, stream>>>(ei, esrc, edst, m, den, z, agg, E, Etot);
  finish_node<<<nodeGrid, 256, 0, stream>>>(agg, b0, g0, be0, nullptr, h, N);

  // ---- Layer 1 (residual) ----
  gat_gemm_wmma<64><<<gemmGrid, 128, 0, stream>>>(h, W1, z, N);
  node_scores<<<nsGrid, 256, 0, stream>>>(z, asrc1, adst1, esrc, edst, N);
  init_buffers<<<initGrid, 256, 0, stream>>>(agg, m, den, N);
  edge_max<<<edgeGrid, 256, 0, stream>>>(ei, esrc, edst, m, E, Etot);
  edge_expsum<<<edgeGrid, 256, 0, stream>>>(ei, esrc, edst, m, den, E, Etot);
  edge_aggregate<<<edgeGrid, 256, 0, stream>>>(ei, esrc, edst, m, den, z, agg, E, Etot);
  finish_node<<<nodeGrid, 256, 0, stream>>>(agg, b1, g1, be1, h, h, N);

  // ---- Output head ----
  out_head<<<nodeGrid, 256, 0, stream>>>(h, Wout, bout, (float*)d_out, N);
}